// Revisit_RDLoss_32847909880461
// MI455X (gfx1250) — compile-verified
//
#include <hip/hip_runtime.h>
#include <hip/hip_bf16.h>
#include <math.h>

// ---------------------------------------------------------------------------
// MI455X (gfx1250) implementation.
// Heavy math = row-softmax over [16, D] + 16x16 Gram via V_WMMA_F32_16X16X4_F32
// accumulated over K = D (up to 1M).  Everything else is streaming reductions.
// No atomics anywhere -> bit-deterministic across replays.
// ---------------------------------------------------------------------------

typedef __attribute__((ext_vector_type(2))) float v2f;
typedef __attribute__((ext_vector_type(8))) float v8f;

#define NBROW  64     // partial blocks per softmax row
#define NBGRAM 512    // blocks for the WMMA gram pass (divides all tile counts)
#define NBREC  512    // blocks for reconstruction-cos pass
#define NBCON  64     // blocks per batch for embedding-cos pass
#define TK     128    // K-tile staged in LDS for the gram pass
#define LDSS   132    // padded LDS row stride (132 % 64 = 4 -> conflict-free)

#define NEG_BIG (-1.0e30f)

// ---------------------------------------------------------------------------
// 1) Per-(row, block) online softmax stats: running max m, running sum exp.
// ---------------------------------------------------------------------------
__global__ __launch_bounds__(256) void stats_kernel(const float* __restrict__ X, int D,
                                                    float* __restrict__ partM,
                                                    float* __restrict__ partS) {
  const int row  = blockIdx.x / NBROW;
  const int part = blockIdx.x % NBROW;
  const int chunk  = D / NBROW;          // divisible for all levels
  const int chunk4 = chunk >> 2;
  const float4* p = reinterpret_cast<const float4*>(X + (size_t)row * D + (size_t)part * chunk);

  float m = NEG_BIG, s = 0.f;
  for (int i = threadIdx.x; i < chunk4; i += 256) {
    float4 v = p[i];
    float lm = fmaxf(fmaxf(v.x, v.y), fmaxf(v.z, v.w));
    float ls = __expf(v.x - lm) + __expf(v.y - lm) + __expf(v.z - lm) + __expf(v.w - lm);
    float mn = fmaxf(m, lm);
    s = s * __expf(m - mn) + ls * __expf(lm - mn);
    m = mn;
  }
  __shared__ float shm[256], shs[256];
  shm[threadIdx.x] = m; shs[threadIdx.x] = s;
  __syncthreads();
  for (int off = 128; off > 0; off >>= 1) {
    if (threadIdx.x < off) {
      float m1 = shm[threadIdx.x], s1 = shs[threadIdx.x];
      float m2 = shm[threadIdx.x + off], s2 = shs[threadIdx.x + off];
      float mn = fmaxf(m1, m2);
      shs[threadIdx.x] = s1 * __expf(m1 - mn) + s2 * __expf(m2 - mn);
      shm[threadIdx.x] = mn;
    }
    __syncthreads();
  }
  if (threadIdx.x == 0) {
    partM[row * NBROW + part] = shm[0];
    partS[row * NBROW + part] = shs[0];
  }
}

__global__ __launch_bounds__(32) void finalize_stats(const float* __restrict__ partM,
                                                     const float* __restrict__ partS,
                                                     float* __restrict__ M,
                                                     float* __restrict__ S) {
  int row = threadIdx.x;
  if (row < 16) {
    float m = NEG_BIG, s = 0.f;
    for (int i = 0; i < NBROW; ++i) {
      float m2 = partM[row * NBROW + i], s2 = partS[row * NBROW + i];
      float mn = fmaxf(m, m2);
      s = s * __expf(m - mn) + s2 * __expf(m2 - mn);
      m = mn;
    }
    M[row] = m; S[row] = s;
  }
}

// ---------------------------------------------------------------------------
// 2) Gram matrix G = P * P^T (P = softmax rows) via V_WMMA_F32_16X16X4_F32.
//    A layout (ISA 7.12.2, 32-bit A 16x4): lanes 0-15 -> K=0,1 ; lanes 16-31
//    -> K=2,3 ; row M = lane&15.  B (4x16) fetch pattern coincides with A
//    (B[k][n] = P[n][k]), so we pass the same v2f twice.  G symmetric -> OK.
// ---------------------------------------------------------------------------
__global__ __launch_bounds__(256) void gram_kernel(const float* __restrict__ X, int D,
                                                   const float* __restrict__ Mrow,
                                                   const float* __restrict__ Srow,
                                                   float* __restrict__ partial) {
  __shared__ float tile[16 * LDSS];
  __shared__ float red[8 * 256];
  __shared__ float shM[16], shSinv[16];

  const int tid  = threadIdx.x;
  const int wave = tid >> 5;
  const int lane = tid & 31;
  const int mrow = lane & 15;
  const int kb   = (lane >> 4) << 1;   // 0 for lanes 0-15, 2 for lanes 16-31

  if (tid < 16) { shM[tid] = Mrow[tid]; shSinv[tid] = 1.0f / Srow[tid]; }

  v8f acc = {};
  const int tiles = D / TK;            // 8192 / 4096 / 2048 -> /NBGRAM exact
  for (int tl = blockIdx.x; tl < tiles; tl += NBGRAM) {
    __syncthreads();                   // prev compute done; shM visible (1st it)
    const int k0 = tl * TK;
    for (int idx = tid; idx < 16 * (TK / 4); idx += 256) {
      int r  = idx >> 5;               // TK/4 == 32
      int c4 = idx & 31;
      float4 v = reinterpret_cast<const float4*>(X + (size_t)r * D + k0)[c4];
      float mr = shM[r], si = shSinv[r];
      float* dst = &tile[r * LDSS + (c4 << 2)];
      dst[0] = __expf(v.x - mr) * si;
      dst[1] = __expf(v.y - mr) * si;
      dst[2] = __expf(v.z - mr) * si;
      dst[3] = __expf(v.w - mr) * si;
    }
    __syncthreads();
    // 32 K-chunks of 4 per tile; wave w owns chunks [4w, 4w+4)
#pragma unroll
    for (int cc = 0; cc < 4; ++cc) {
      int k = ((wave << 2) + cc) << 2;
      v2f a;
      a.x = tile[mrow * LDSS + k + kb];
      a.y = tile[mrow * LDSS + k + kb + 1];
      acc = __builtin_amdgcn_wmma_f32_16x16x4_f32(false, a, false, a,
                                                  (short)0, acc, false, false);
    }
  }
  __syncthreads();
  {
    const int Mi = (lane >> 4) << 3;   // C/D layout: VGPR r -> M = r (+8 hi half)
    const int Ni = lane & 15;
#pragma unroll
    for (int r = 0; r < 8; ++r) red[wave * 256 + (Mi + r) * 16 + Ni] = acc[r];
  }
  __syncthreads();
  float s = 0.f;
#pragma unroll
  for (int w = 0; w < 8; ++w) s += red[w * 256 + tid];
  partial[(size_t)blockIdx.x * 256 + tid] = s;
}

__global__ __launch_bounds__(256) void gram_reduce(const float* __restrict__ part,
                                                   float* __restrict__ outG) {
  float s = 0.f;
  for (int b = 0; b < NBGRAM; ++b) s += part[b * 256 + threadIdx.x];
  outG[threadIdx.x] = s;
}

// ---------------------------------------------------------------------------
// 3) Reconstruction cosine: per (b,h,w) dot/norm over channels, sum(1-cos).
// ---------------------------------------------------------------------------
__global__ __launch_bounds__(256) void rec_kernel(const float* __restrict__ A,
                                                  const float* __restrict__ P,
                                                  int C, int HW,
                                                  float* __restrict__ partial) {
  const int total4 = (16 * HW) >> 2;
  const int hw4b   = HW >> 2;
  float acc = 0.f;
  for (int p4 = blockIdx.x * 256 + threadIdx.x; p4 < total4; p4 += NBREC * 256) {
    int b   = p4 / hw4b;
    int hw4 = p4 - b * hw4b;
    const float4* a4 = reinterpret_cast<const float4*>(A + (size_t)b * C * HW) + hw4;
    const float4* q4 = reinterpret_cast<const float4*>(P + (size_t)b * C * HW) + hw4;
    float dx=0,dy=0,dz=0,dw=0, ax=0,ay=0,az=0,aw=0, px=0,py=0,pz=0,pw=0;
    for (int c = 0; c < C; ++c) {
      float4 av = a4[(size_t)c * hw4b];
      float4 pv = q4[(size_t)c * hw4b];
      dx = fmaf(av.x, pv.x, dx); ax = fmaf(av.x, av.x, ax); px = fmaf(pv.x, pv.x, px);
      dy = fmaf(av.y, pv.y, dy); ay = fmaf(av.y, av.y, ay); py = fmaf(pv.y, pv.y, py);
      dz = fmaf(av.z, pv.z, dz); az = fmaf(av.z, av.z, az); pz = fmaf(pv.z, pv.z, pz);
      dw = fmaf(av.w, pv.w, dw); aw = fmaf(av.w, av.w, aw); pw = fmaf(pv.w, pv.w, pw);
    }
    acc += 1.f - dx / fmaxf(sqrtf(ax) * sqrtf(px), 1e-8f);
    acc += 1.f - dy / fmaxf(sqrtf(ay) * sqrtf(py), 1e-8f);
    acc += 1.f - dz / fmaxf(sqrtf(az) * sqrtf(pz), 1e-8f);
    acc += 1.f - dw / fmaxf(sqrtf(aw) * sqrtf(pw), 1e-8f);
  }
  __shared__ float sh[256];
  sh[threadIdx.x] = acc;
  __syncthreads();
  for (int off = 128; off > 0; off >>= 1) {
    if (threadIdx.x < off) sh[threadIdx.x] += sh[threadIdx.x + off];
    __syncthreads();
  }
  if (threadIdx.x == 0) partial[blockIdx.x] = sh[0];
}

// ---------------------------------------------------------------------------
// 4) Embedding cosine partials: per (b, chunk) -> (dot, |x|^2, |y|^2).
// ---------------------------------------------------------------------------
__global__ __launch_bounds__(256) void con_kernel(const float* __restrict__ X,
                                                  const float* __restrict__ Y,
                                                  int N, float* __restrict__ partial) {
  const int b = blockIdx.x / NBCON, part = blockIdx.x % NBCON;
  const int chunk  = N / NBCON;
  const int chunk4 = chunk >> 2;
  const float4* x4 = reinterpret_cast<const float4*>(X + (size_t)b * N + (size_t)part * chunk);
  const float4* y4 = reinterpret_cast<const float4*>(Y + (size_t)b * N + (size_t)part * chunk);
  float dot = 0.f, nx = 0.f, ny = 0.f;
  for (int i = threadIdx.x; i < chunk4; i += 256) {
    float4 a = x4[i], c = y4[i];
    dot = fmaf(a.x, c.x, fmaf(a.y, c.y, fmaf(a.z, c.z, fmaf(a.w, c.w, dot))));
    nx  = fmaf(a.x, a.x, fmaf(a.y, a.y, fmaf(a.z, a.z, fmaf(a.w, a.w, nx))));
    ny  = fmaf(c.x, c.x, fmaf(c.y, c.y, fmaf(c.z, c.z, fmaf(c.w, c.w, ny))));
  }
  __shared__ float sh[3 * 256];
  sh[threadIdx.x] = dot; sh[256 + threadIdx.x] = nx; sh[512 + threadIdx.x] = ny;
  __syncthreads();
  for (int off = 128; off > 0; off >>= 1) {
    if (threadIdx.x < off) {
      sh[threadIdx.x]       += sh[threadIdx.x + off];
      sh[256 + threadIdx.x] += sh[256 + threadIdx.x + off];
      sh[512 + threadIdx.x] += sh[512 + threadIdx.x + off];
    }
    __syncthreads();
  }
  if (threadIdx.x == 0) {
    float* o = partial + (size_t)blockIdx.x * 3;
    o[0] = sh[0]; o[1] = sh[256]; o[2] = sh[512];
  }
}

// ---------------------------------------------------------------------------
// 5) Final: Sinkhorn eps-annealing on 16x16 matrices + combine all losses.
// ---------------------------------------------------------------------------
__device__ __forceinline__ float softmin16(float eps, const float* C, int base,
                                           int stride, const float* h, float logw) {
  float m = NEG_BIG, s = 0.f;
  for (int j = 0; j < 16; ++j) {
    float t = logw + (h[j] - C[base + j * stride]) / eps;
    float mn = fmaxf(m, t);
    s = s * __expf(m - mn) + __expf(t - mn);
    m = mn;
  }
  return -eps * (m + logf(s));
}

__global__ __launch_bounds__(64) void final_kernel(const float* __restrict__ gram,
                                                   const int* __restrict__ shuffle,
                                                   const float* __restrict__ recPart,
                                                   const float* __restrict__ conPart,
                                                   float* __restrict__ out) {
  __shared__ float Cxy[256], Cxx[256], Cyy[256];
  __shared__ float f[16], g[16], fx[16], gy[16];
  __shared__ float ft[16], gt[16], fxt[16], gyt[16];
  __shared__ int   sidx[16];
  __shared__ float dd[16];
  const float logw = -2.772588722239781f;   // -ln(16)
  const int t = threadIdx.x;
  float ssot = 0.f;

  if (t < 16) sidx[t] = shuffle[t];
  __syncthreads();

  for (int l = 0; l < 3; ++l) {
    const float* G = gram + l * 256;
    if (t < 16) { dd[t] = G[t * 17]; f[t] = 0.f; g[t] = 0.f; fx[t] = 0.f; gy[t] = 0.f; }
    __syncthreads();
    for (int idx = t; idx < 256; idx += 64) {
      int i = idx >> 4, j = idx & 15;
      int si = sidx[i], sj = sidx[j];
      Cxy[idx] = 0.5f * fmaxf(dd[i]  + dd[sj] - 2.f * G[i * 16 + sj], 0.f);
      Cxx[idx] = 0.5f * fmaxf(dd[i]  + dd[j]  - 2.f * G[i * 16 + j],  0.f);
      Cyy[idx] = 0.5f * fmaxf(dd[si] + dd[sj] - 2.f * G[si * 16 + sj], 0.f);
    }
    __syncthreads();

    float e = 4.0f;                          // diameter^2
    for (int k = 0; k < 73; ++k) {           // eps schedule: max(4*0.9025^k, blur^2)
      float eps = fmaxf(e, 0.0025f);
      if (t < 16) {
        ft[t]  = softmin16(eps, Cxy, t * 16, 1,  g,  logw);
        gt[t]  = softmin16(eps, Cxy, t,      16, f,  logw);
        fxt[t] = softmin16(eps, Cxx, t * 16, 1,  fx, logw);
        gyt[t] = softmin16(eps, Cyy, t * 16, 1,  gy, logw);
      }
      __syncthreads();
      if (t < 16) {
        f[t]  = 0.5f * (f[t]  + ft[t]);
        g[t]  = 0.5f * (g[t]  + gt[t]);
        fx[t] = 0.5f * (fx[t] + fxt[t]);
        gy[t] = 0.5f * (gy[t] + gyt[t]);
      }
      __syncthreads();
      e *= 0.9025f;
    }
    const float epsT = 0.0025f;              // final non-averaged extrapolation
    if (t < 16) {
      ft[t] = softmin16(epsT, Cxy, t * 16, 1,  g, logw);
      gt[t] = softmin16(epsT, Cxy, t,      16, f, logw);
    }
    __syncthreads();
    if (t == 0) {
      float a = 0.f, b = 0.f;
      for (int i = 0; i < 16; ++i) { a += ft[i] - fx[i]; b += gt[i] - gy[i]; }
      ssot += (a + b) * (1.0f / 16.0f);
    }
    __syncthreads();
  }

  if (t == 0) {
    const int HWs[3] = {4096, 1024, 256};
    float rec = 0.f;
    for (int l = 0; l < 3; ++l) {
      float s = 0.f;
      for (int b = 0; b < NBREC; ++b) s += recPart[l * NBREC + b];
      rec += s / (16.0f * (float)HWs[l]);
    }
    float con = 0.f;
    for (int l = 0; l < 3; ++l) {
      float cl = 0.f;
      for (int b = 0; b < 16; ++b) {
        float dot = 0.f, nx = 0.f, ny = 0.f;
        const float* p = conPart + ((size_t)(l * 16 + b) * NBCON) * 3;
        for (int q = 0; q < NBCON; ++q) { dot += p[q*3]; nx += p[q*3+1]; ny += p[q*3+2]; }
        float cosv = dot / fmaxf(sqrtf(nx) * sqrtf(ny), 1e-8f);
        cl += fmaxf(cosv + 0.5f, 0.f);       // margin = -0.5, target = -1
      }
      con += cl * (1.0f / 16.0f);
    }
    out[0] = (ssot + 0.01f * rec + 0.1f * con) * (1.0f / 1.11f);
  }
}

// ---------------------------------------------------------------------------
extern "C" void kernel_launch(void* const* d_in, const int* in_sizes, int n_in,
                              void* d_out, int out_size, void* d_ws, size_t ws_size,
                              hipStream_t stream) {
  (void)in_sizes; (void)n_in; (void)out_size; (void)ws_size;
  const float* noised[3] = {(const float*)d_in[0], (const float*)d_in[1], (const float*)d_in[2]};
  const float* pnoise[3] = {(const float*)d_in[3], (const float*)d_in[4], (const float*)d_in[5]};
  const float* pnorm[3]  = {(const float*)d_in[6], (const float*)d_in[7], (const float*)d_in[8]};
  const int*   shuffle   = (const int*)d_in[9];
  float* out = (float*)d_out;
  float* ws  = (float*)d_ws;

  const int Cs[3]  = {256, 512, 1024};
  const int HWs[3] = {4096, 1024, 256};

  // workspace layout (floats); every slot is written before it is read.
  float* gramPart = ws;                          // NBGRAM*256
  float* partM    = gramPart + NBGRAM * 256;     // 16*NBROW (reused per level)
  float* partS    = partM + 16 * NBROW;          // 16*NBROW
  float* Mrow     = partS + 16 * NBROW;          // 3*16
  float* Srow     = Mrow + 48;                   // 3*16
  float* gramF    = Srow + 48;                   // 3*256
  float* recP     = gramF + 768;                 // 3*NBREC
  float* conP     = recP + 3 * NBREC;            // 3*16*NBCON*3

  for (int l = 0; l < 3; ++l) {
    const int D = Cs[l] * HWs[l];
    stats_kernel<<<16 * NBROW, 256, 0, stream>>>(pnorm[l], D, partM, partS);
    finalize_stats<<<1, 32, 0, stream>>>(partM, partS, Mrow + l * 16, Srow + l * 16);
    gram_kernel<<<NBGRAM, 256, 0, stream>>>(pnorm[l], D, Mrow + l * 16, Srow + l * 16, gramPart);
    gram_reduce<<<1, 256, 0, stream>>>(gramPart, gramF + l * 256);
    rec_kernel<<<NBREC, 256, 0, stream>>>(pnoise[l], pnorm[l], Cs[l], HWs[l], recP + l * NBREC);
    con_kernel<<<16 * NBCON, 256, 0, stream>>>(noised[l], pnorm[l], D, conP + (size_t)l * 16 * NBCON * 3);
  }
  final_kernel<<<1, 64, 0, stream>>>(gramF, shuffle, recP, conP, out);
}